// close_27066883899534
// MI455X (gfx1250) — compile-verified
//
#include <hip/hip_runtime.h>

#define BSZ 32
#define TT  12
#define CC  2
#define NN  1024
#define KK  10
#define DD  (CC*TT)   // 24

typedef __attribute__((ext_vector_type(2))) float v2f;
typedef __attribute__((ext_vector_type(8))) float v8f;

// ---------------------------------------------------------------------------
// Kernel 1: fnT[b][d][n] = x_c[b,t,c,n] / (||feats[b,n,:]|| + 1e-8), d=c*T+t
// Coalesced across n for both loads and stores.
// ---------------------------------------------------------------------------
__global__ __launch_bounds__(256)
void fnorm_kernel(const float* __restrict__ x, float* __restrict__ fnT) {
    int tid = blockIdx.x * blockDim.x + threadIdx.x;   // 0 .. BSZ*NN-1
    int b = tid >> 10;
    int n = tid & (NN - 1);
    float f[DD];
    float ss = 0.0f;
#pragma unroll
    for (int c = 0; c < CC; ++c)
#pragma unroll
        for (int t = 0; t < TT; ++t) {
            float v = x[((b * TT + t) * CC + c) * NN + n];
            f[c * TT + t] = v;
            ss += v * v;
        }
    float inv = 1.0f / (sqrtf(ss) + 1e-8f);
#pragma unroll
    for (int d = 0; d < DD; ++d)
        fnT[(b * DD + d) * NN + n] = f[d] * inv;
}

// ---------------------------------------------------------------------------
// Kernel 2: one wave handles 32 rows of one batch.
//  - adj tiles via V_WMMA_F32_16X16X4_F32 (6 chained K=4 steps, fp32 exact)
//  - streaming per-row top-10 (register insertion sort, early-out on min)
//  - fused gather + weighted sum -> out[b,n,t]
// ---------------------------------------------------------------------------
__global__ __launch_bounds__(32)
void adj_topk_out_kernel(const float* __restrict__ x,
                         const float* __restrict__ fnT,
                         const float* __restrict__ W,
                         const int* __restrict__ flowp,
                         float* __restrict__ out) {
    __shared__ float tile[2 * 16 * 16];

    const int lane = threadIdx.x;            // 0..31
    const int b    = blockIdx.x >> 5;        // 32 row-groups per batch
    const int rg   = blockIdx.x & 31;
    const int rowbase = rg * 32;

    const int half = lane >> 4;              // 0 or 1
    const int lm   = lane & 15;
    const int koff = half * 2;               // lanes 0-15: K=0,1 ; lanes 16-31: K=2,3

    const float* fb = fnT + b * (DD * NN);

    // A fragments for our 32 rows (two 16-row tiles), loaded once.
    v2f A0[6], A1[6];
#pragma unroll
    for (int s = 0; s < 6; ++s) {
        int d0 = 4 * s + koff;
        v2f a;
        a[0] = fb[d0 * NN + rowbase + lm];
        a[1] = fb[(d0 + 1) * NN + rowbase + lm];
        A0[s] = a;
        a[0] = fb[d0 * NN + rowbase + 16 + lm];
        a[1] = fb[(d0 + 1) * NN + rowbase + 16 + lm];
        A1[s] = a;
    }

    // Per-lane top-10 (lane owns row rowbase+lane), sorted descending.
    float vals[KK];
    int   idx[KK];
#pragma unroll
    for (int j = 0; j < KK; ++j) { vals[j] = -3.0e38f; idx[j] = 0; }
    float vmin = -3.0e38f;

    for (int ct = 0; ct < 64; ++ct) {
        const int cbase = ct * 16;
        v8f acc0 = {};
        v8f acc1 = {};
#pragma unroll
        for (int s = 0; s < 6; ++s) {
            int d0 = 4 * s + koff;
            v2f Bf;
            Bf[0] = fb[d0 * NN + cbase + lm];
            Bf[1] = fb[(d0 + 1) * NN + cbase + lm];
            acc0 = __builtin_amdgcn_wmma_f32_16x16x4_f32(
                false, A0[s], false, Bf, (short)0, acc0, false, false);
            acc1 = __builtin_amdgcn_wmma_f32_16x16x4_f32(
                false, A1[s], false, Bf, (short)0, acc1, false, false);
        }
        // C/D layout: lane holds column N=lm, rows M = half*8 + v
#pragma unroll
        for (int v = 0; v < 8; ++v) {
            tile[0 * 256 + lm * 16 + half * 8 + v] = acc0[v];
            tile[1 * 256 + lm * 16 + half * 8 + v] = acc1[v];
        }
        __syncthreads();

        // Lane owns row rowbase + half*16 + lm == rowbase + lane
#pragma unroll
        for (int c = 0; c < 16; ++c) {
            float v = tile[half * 256 + c * 16 + lm];
            int gi = cbase + c;
            if (v > vmin) {                 // early-out: below current 10th
                float cv = v; int ci = gi;  // bubble-insert (strict >, keeps
#pragma unroll                              // lower index first on ties)
                for (int j = 0; j < KK; ++j) {
                    bool gt = cv > vals[j];
                    float tv = vals[j]; int ti = idx[j];
                    vals[j] = gt ? cv : tv;
                    idx[j]  = gt ? ci : ti;
                    cv = gt ? tv : cv;
                    ci = gt ? ti : ci;
                }
                vmin = vals[KK - 1];
            }
        }
        __syncthreads();
    }

    // Fused gather + weighted sum:
    // out[b,n,t] = (W0+1)*sxf[b,n,t] + sum_k W[k+1]*sxf[b, idx[k], t]
    const int fl = flowp[0];
    const int n  = rowbase + lane;
    float w[KK + 1];
#pragma unroll
    for (int j = 0; j <= KK; ++j) w[j] = W[j];
    const float w0p1 = w[0] + 1.0f;
#pragma unroll
    for (int t = 0; t < TT; ++t) {
        const float* xt = x + ((b * TT + t) * CC + fl) * NN;
        float s = w0p1 * xt[n];
#pragma unroll
        for (int k = 0; k < KK; ++k) s += w[k + 1] * xt[idx[k]];
        out[(b * NN + n) * TT + t] = s;
    }
}

// ---------------------------------------------------------------------------
extern "C" void kernel_launch(void* const* d_in, const int* in_sizes, int n_in,
                              void* d_out, int out_size, void* d_ws, size_t ws_size,
                              hipStream_t stream) {
    const float* x    = (const float*)d_in[0];   // (32,12,2,1024) f32
    const float* W    = (const float*)d_in[1];   // (11,) f32
    const int*   flow = (const int*)d_in[2];     // scalar
    float*       out  = (float*)d_out;           // (32,1024,12) f32
    float*       fnT  = (float*)d_ws;            // 32*24*1024 f32 = 3 MB

    fnorm_kernel<<<(BSZ * NN) / 256, 256, 0, stream>>>(x, fnT);
    adj_topk_out_kernel<<<BSZ * (NN / 32), 32, 0, stream>>>(x, fnT, W, flow, out);
}